// DepthwiseDilation2D_2551210573877
// MI455X (gfx1250) — compile-verified
//
#include <hip/hip_runtime.h>
#include <cstdint>
#include <cstddef>

// Problem constants (match the reference).
#define Bn   4
#define Hn   256
#define Wn   256
#define Cn   64
#define DMn  2

// Tile shape per block.
#define TH      8
#define TW      16
#define HALO_H  (TH + 2)
#define HALO_W  (TW + 2)

typedef float v2f __attribute__((ext_vector_type(2)));
typedef float v4f __attribute__((ext_vector_type(4)));
typedef int   v4i __attribute__((ext_vector_type(4)));

#if defined(__HIP_DEVICE_COMPILE__) && __has_builtin(__builtin_amdgcn_global_load_async_to_lds_b128)
#define USE_ASYNC_LDS 1
#else
#define USE_ASYNC_LDS 0
#endif

__global__ __launch_bounds__(256) void
DepthwiseDilation2D_kernel(const float* __restrict__ x,
                           const float* __restrict__ kern,
                           float* __restrict__ out)
{
    // Halo tile: (TH+2) x (TW+2) x C floats = 45 KB of the 320 KB LDS.
    __shared__ __align__(16) float tile[HALO_H * HALO_W * Cn];

    const int tid = threadIdx.x;
    const int bx  = blockIdx.x;          // W tile index
    const int by  = blockIdx.y;          // H tile index
    const int b   = blockIdx.z;          // batch

    const int x0 = bx * TW;
    const int y0 = by * TH;

    const float NEG_INF = -__builtin_inff();

    // ---------------- Stage halo tile into LDS ----------------
    // 180 halo pixels * 16 channel-groups-of-4 = 2880 B128 transfers.
    const int nChunks = HALO_H * HALO_W * (Cn / 4);
    for (int t = tid; t < nChunks; t += 256) {
        const int cg = t & (Cn / 4 - 1);     // channel group (4 floats)
        const int p  = t >> 4;               // halo pixel 0..179
        const int ly = p / HALO_W;
        const int lx = p - ly * HALO_W;
        const int gy = y0 + ly - 1;
        const int gx = x0 + lx - 1;
        const int ldsOff = (ly * HALO_W + lx) * Cn + cg * 4;   // float index
        if ((unsigned)gy < (unsigned)Hn && (unsigned)gx < (unsigned)Wn) {
            const float* gsrc =
                x + ((((size_t)b * Hn + gy) * Wn + gx) * Cn + cg * 4);
#if USE_ASYNC_LDS
            // CDNA5 async DMA: global -> LDS without touching VGPRs (ASYNCcnt).
            auto g1 = (__attribute__((address_space(1))) v4i*)(uintptr_t)gsrc;
            auto l3 = (__attribute__((address_space(3))) v4i*)
                          (uint32_t)(uintptr_t)(&tile[ldsOff]);
            __builtin_amdgcn_global_load_async_to_lds_b128(g1, l3, 0, 0);
#else
            *reinterpret_cast<v4f*>(&tile[ldsOff]) =
                *reinterpret_cast<const v4f*>(gsrc);
#endif
        } else {
            v4f ninf = {NEG_INF, NEG_INF, NEG_INF, NEG_INF};
            *reinterpret_cast<v4f*>(&tile[ldsOff]) = ninf;  // disjoint from DMA targets
        }
    }
#if USE_ASYNC_LDS
#if __has_builtin(__builtin_amdgcn_s_wait_asynccnt)
    __builtin_amdgcn_s_wait_asynccnt(0);
#else
    asm volatile("s_wait_asynccnt 0" ::: "memory");
#endif
#endif
    __syncthreads();

    // ---------------- Compute ----------------
    // wave32 mapping: lane -> channel pair, wave -> output row.
    const int lane = tid & 31;
    const int row  = tid >> 5;           // 0..7  (TH rows)
    const int c2   = lane * 2;           // channels c2, c2+1

    // Per-thread taps: kern[((i*3+j)*Cn + c)*DMn + di]; float4 = {kA0,kA1,kB0,kB1}.
    v4f kw[9];
#pragma unroll
    for (int ij = 0; ij < 9; ++ij)
        kw[ij] = *reinterpret_cast<const v4f*>(kern + ((size_t)ij * Cn + c2) * DMn);

    // Rolling 3-column window; each column = rows (row..row+2) as float2 (2 channels).
    const int rowBase = row * HALO_W * Cn + c2;
    auto ldcol = [&](int lx, v2f* col) {
        const int base = rowBase + lx * Cn;
        col[0] = *reinterpret_cast<const v2f*>(&tile[base]);
        col[1] = *reinterpret_cast<const v2f*>(&tile[base + HALO_W * Cn]);
        col[2] = *reinterpret_cast<const v2f*>(&tile[base + 2 * HALO_W * Cn]);
    };

    v2f w0[3], w1[3], w2[3];
    ldcol(0, w0);
    ldcol(1, w1);

    float* outRow = out + (((size_t)b * Hn + (y0 + row)) * Wn + x0) * (DMn * Cn);

    for (int xo = 0; xo < TW; ++xo) {
        ldcol(xo + 2, w2);

        float m00 = NEG_INF, m01 = NEG_INF;   // channel c2   : dm0, dm1
        float m10 = NEG_INF, m11 = NEG_INF;   // channel c2+1 : dm0, dm1
#pragma unroll
        for (int i = 0; i < 3; ++i) {
            const v4f kA = kw[i * 3 + 0]; const v2f vA = w0[i];
            m00 = fmaxf(m00, vA.x + kA.x); m01 = fmaxf(m01, vA.x + kA.y);
            m10 = fmaxf(m10, vA.y + kA.z); m11 = fmaxf(m11, vA.y + kA.w);
            const v4f kB = kw[i * 3 + 1]; const v2f vB = w1[i];
            m00 = fmaxf(m00, vB.x + kB.x); m01 = fmaxf(m01, vB.x + kB.y);
            m10 = fmaxf(m10, vB.y + kB.z); m11 = fmaxf(m11, vB.y + kB.w);
            const v4f kC = kw[i * 3 + 2]; const v2f vC = w2[i];
            m00 = fmaxf(m00, vC.x + kC.x); m01 = fmaxf(m01, vC.x + kC.y);
            m10 = fmaxf(m10, vC.y + kC.z); m11 = fmaxf(m11, vC.y + kC.w);
        }

        // Output layout: channel index di*C + c. Write-once -> non-temporal.
        float* op = outRow + (size_t)xo * (DMn * Cn);
        v2f o0 = {m00, m10};     // dm = 0, channels c2, c2+1
        v2f o1 = {m01, m11};     // dm = 1
        __builtin_nontemporal_store(o0, reinterpret_cast<v2f*>(op + c2));
        __builtin_nontemporal_store(o1, reinterpret_cast<v2f*>(op + Cn + c2));

        // Slide window.
        w0[0] = w1[0]; w0[1] = w1[1]; w0[2] = w1[2];
        w1[0] = w2[0]; w1[1] = w2[1]; w1[2] = w2[2];
    }
}

extern "C" void kernel_launch(void* const* d_in, const int* in_sizes, int n_in,
                              void* d_out, int out_size, void* d_ws, size_t ws_size,
                              hipStream_t stream) {
    (void)in_sizes; (void)n_in; (void)out_size; (void)d_ws; (void)ws_size;
    const float* x    = (const float*)d_in[0];
    const float* kern = (const float*)d_in[1];
    float*       out  = (float*)d_out;

    dim3 grid(Wn / TW, Hn / TH, Bn);   // 16 x 32 x 4 blocks
    dim3 block(256);                   // 8 waves (wave32)
    DepthwiseDilation2D_kernel<<<grid, block, 0, stream>>>(x, kern, out);
}